// LinearAttentionTransformer_15444702396584
// MI455X (gfx1250) — compile-verified
//
#include <hip/hip_runtime.h>

typedef __attribute__((ext_vector_type(2))) float v2f;
typedef __attribute__((ext_vector_type(8))) float v8f;

#define N_FIRMS 8192
#define DIM 128
#define GRAM_WGS 16
#define ROWS_PER_WG (N_FIRMS / GRAM_WGS) /* 512 */

// ---------------------------------------------------------------------------
// Kernel 1: per-workgroup partial Gram matrix  P_wg = X_chunk^T X_chunk
// Uses V_WMMA_F32_16X16X4_F32 as a rank-4 update; 8 waves cover the 8 tile
// rows of the 128x128 output, each wave does 8 tiles (A-fragment reused).
// ---------------------------------------------------------------------------
__global__ __launch_bounds__(256) void gram_kernel(const float* __restrict__ X,
                                                   float* __restrict__ partials) {
  __shared__ __align__(16) float xs[4 * DIM];  // one 4x128 row block of X

  const int tid   = threadIdx.x;
  const int wave  = tid >> 5;       // 0..7  -> tile row (a0 = 16*wave)
  const int lane  = tid & 31;
  const int laneM = lane & 15;      // M (for A) / N (for B) within tile
  const int hi    = lane >> 4;      // 0: K=0,1   1: K=2,3  (ISA 16x4 A layout)
  const int a0    = wave * 16;

  v8f acc[8] = {};                  // 8 tiles x 8 VGPRs of f32 accumulator

  const float* Xw = X + (size_t)blockIdx.x * ROWS_PER_WG * DIM;

  for (int ks = 0; ks < ROWS_PER_WG; ks += 4) {
    __syncthreads();                          // protect LDS reuse
    xs[tid]       = Xw[ks * DIM + tid];       // rows ks, ks+1
    xs[tid + 256] = Xw[ks * DIM + tid + 256]; // rows ks+2, ks+3
    if (ks + 4 < ROWS_PER_WG)                 // gfx1250 global_prefetch_b8
      __builtin_prefetch(&Xw[(ks + 4) * DIM + tid * 2], 0, 0);
    __syncthreads();

    // A fragment (16x4, f32): lane half selects K pair; shared across 8 tiles
    v2f a;
    a.x = xs[(2 * hi + 0) * DIM + a0 + laneM];   // K = 0 or 2
    a.y = xs[(2 * hi + 1) * DIM + a0 + laneM];   // K = 1 or 3

#pragma unroll
    for (int tc = 0; tc < 8; ++tc) {
      // B fragment (4x16, f32): same lane pattern, column base = 16*tc
      v2f b;
      b.x = xs[(2 * hi + 0) * DIM + tc * 16 + laneM];
      b.y = xs[(2 * hi + 1) * DIM + tc * 16 + laneM];
      acc[tc] = __builtin_amdgcn_wmma_f32_16x16x4_f32(
          /*neg_a=*/false, a, /*neg_b=*/false, b,
          /*c_mod=*/(short)0, acc[tc], /*reuse_a=*/false, /*reuse_b=*/false);
    }
  }

  // C/D layout: VGPR r -> M = r (lanes 0-15) or M = r+8 (lanes 16-31), N = laneM
  float* P = partials + (size_t)blockIdx.x * DIM * DIM;
#pragma unroll
  for (int tc = 0; tc < 8; ++tc) {
#pragma unroll
    for (int r = 0; r < 8; ++r) {
      const int row = a0 + r + hi * 8;
      const int col = tc * 16 + laneM;
      P[row * DIM + col] = acc[tc][r];
    }
  }
}

// ---------------------------------------------------------------------------
// Kernel 2: fold the whole network into one 128-vector:
//   w_eff = Wq^T * Wk * (sum_p P_p) * Wv^T * head_w^T
// Single 128-thread block; mat-vec chain, G never materialized.
// ---------------------------------------------------------------------------
__global__ __launch_bounds__(128) void weff_kernel(const float* __restrict__ partials,
                                                   const float* __restrict__ Wq,
                                                   const float* __restrict__ Wk,
                                                   const float* __restrict__ Wv,
                                                   const float* __restrict__ head_w,
                                                   float* __restrict__ w_eff) {
  __shared__ float t1[DIM];
  __shared__ float t2[DIM];
  __shared__ float t3[DIM];
  const int d = threadIdx.x;

  // t1 = Wv^T h    (t1[d] = sum_e Wv[e,d] * h[e])
  float s = 0.f;
  for (int e = 0; e < DIM; ++e) s += Wv[e * DIM + d] * head_w[e];
  t1[d] = s;
  __syncthreads();

  // t2 = G t1 = sum_p P_p t1
  s = 0.f;
  for (int p = 0; p < GRAM_WGS; ++p) {
    const float* P = partials + (size_t)p * DIM * DIM + (size_t)d * DIM;
    float sp = 0.f;
    for (int b = 0; b < DIM; ++b) sp += P[b] * t1[b];
    s += sp;
  }
  t2[d] = s;
  __syncthreads();

  // t3 = Wk t2     (t3[e] = sum_q Wk[e,q] * t2[q])
  s = 0.f;
  for (int q = 0; q < DIM; ++q) s += Wk[d * DIM + q] * t2[q];
  t3[d] = s;
  __syncthreads();

  // w_eff = Wq^T t3
  s = 0.f;
  for (int e = 0; e < DIM; ++e) s += Wq[e * DIM + d] * t3[e];
  w_eff[d] = s;
}

// ---------------------------------------------------------------------------
// Kernel 3: preds = X * w_eff + b.  4 lanes per row, float4 loads,
// wave32 shfl_xor reduction. 64 rows per 256-thread block.
// ---------------------------------------------------------------------------
__global__ __launch_bounds__(256) void preds_kernel(const float* __restrict__ X,
                                                    const float* __restrict__ w_eff,
                                                    const float* __restrict__ head_b,
                                                    float* __restrict__ out) {
  __shared__ __align__(16) float w[DIM];
  const int tid = threadIdx.x;
  if (tid < DIM) w[tid] = w_eff[tid];
  __syncthreads();

  const int sub = tid & 3;                 // quarter of the row
  const int r   = tid >> 2;                // 0..63
  const int row = blockIdx.x * 64 + r;

  const float4* Xp = (const float4*)(X + (size_t)row * DIM + sub * 32);
  const float4* wp = (const float4*)(w + sub * 32);

  float s = 0.f;
#pragma unroll
  for (int j = 0; j < 8; ++j) {
    const float4 x  = Xp[j];
    const float4 ww = wp[j];
    s += x.x * ww.x + x.y * ww.y + x.z * ww.z + x.w * ww.w;
  }
  s += __shfl_xor(s, 1, 32);
  s += __shfl_xor(s, 2, 32);
  if (sub == 0) out[row] = s + head_b[0];
}

// ---------------------------------------------------------------------------
extern "C" void kernel_launch(void* const* d_in, const int* in_sizes, int n_in,
                              void* d_out, int out_size, void* d_ws, size_t ws_size,
                              hipStream_t stream) {
  const float* X      = (const float*)d_in[0];
  const float* Wq     = (const float*)d_in[1];
  const float* Wk     = (const float*)d_in[2];
  const float* Wv     = (const float*)d_in[3];
  const float* head_w = (const float*)d_in[4];
  const float* head_b = (const float*)d_in[5];
  float* out = (float*)d_out;

  float* partials = (float*)d_ws;                         // 16 * 128*128 f32
  float* w_eff    = partials + (size_t)GRAM_WGS * DIM * DIM;  // 128 f32

  gram_kernel<<<GRAM_WGS, 256, 0, stream>>>(X, partials);
  weff_kernel<<<1, 128, 0, stream>>>(partials, Wq, Wk, Wv, head_w, w_eff);
  preds_kernel<<<N_FIRMS / 64, 256, 0, stream>>>(X, w_eff, head_b, out);
}